// PytorchModel_2078764171677
// MI455X (gfx1250) — compile-verified
//
#include <hip/hip_runtime.h>
#include <hip/hip_bf16.h>
#include <stdint.h>

// Depthwise causal conv1d single-token step + SiLU (Mamba conv step).
//   y[d] = silu( s[d,1]*w[d,0] + s[d,2]*w[d,1] + s[d,3]*w[d,2] + x[d]*w[d,3] )
// D = 10240 channels, K = 4 taps. Pure memory/latency-bound (~200 KB traffic,
// ~9 ns at 23.3 TB/s) -> one small wave32 kernel, 128-bit coalesced loads,
// free bf16->f32 conversion via shifts, fp32 accumulate, packed b32 store.

#ifndef CONV_D
#define CONV_D 10240
#endif
#define CONV_K 4
#define VEC 2        // channels per thread
#define BLOCK 256    // 8 wave32 waves per block

// bf16 (stored in low/high half of a u32 pair) -> f32 : exact, shift only.
__device__ __forceinline__ float bf16_lo(uint32_t u) {
    return __builtin_bit_cast(float, u << 16);
}
__device__ __forceinline__ float bf16_hi(uint32_t u) {
    return __builtin_bit_cast(float, u & 0xffff0000u);
}

__device__ __forceinline__ uint16_t f32_to_bf16_bits(float f) {
    __hip_bfloat16 h = __float2bfloat16(f);  // round-to-nearest-even
    return __builtin_bit_cast(uint16_t, h);
}

__device__ __forceinline__ float silu(float y) {
    // y * sigmoid(y); v_exp_f32 + v_rcp path
    return y * (1.0f / (1.0f + __expf(-y)));
}

__global__ __launch_bounds__(BLOCK) void conv_step_silu_kernel(
    const __hip_bfloat16* __restrict__ x,      // (D)
    const __hip_bfloat16* __restrict__ w,      // (D, K) row-major
    const __hip_bfloat16* __restrict__ st,     // (D, K) row-major
    __hip_bfloat16* __restrict__ out)          // (D)
{
    const int d0 = (blockIdx.x * BLOCK + threadIdx.x) * VEC;
    if (d0 >= CONV_D) return;

    // 2 channels * 4 taps * 2B = 16B -> single global_load_b128 each.
    const uint4 s  = *reinterpret_cast<const uint4*>(st + (size_t)d0 * CONV_K);
    const uint4 wv = *reinterpret_cast<const uint4*>(w  + (size_t)d0 * CONV_K);
    const uint32_t xs = *reinterpret_cast<const uint32_t*>(x + d0);

    // Channel d0: taps (s.x lo,hi | s.y lo,hi) = s0,s1,s2,s3 ; weights in wv.x/wv.y
    // shifted window = [s1, s2, s3, x0]
    float y0 = bf16_hi(s.x) * bf16_lo(wv.x);
    y0 = fmaf(bf16_lo(s.y), bf16_hi(wv.x), y0);
    y0 = fmaf(bf16_hi(s.y), bf16_lo(wv.y), y0);
    y0 = fmaf(bf16_lo(xs),  bf16_hi(wv.y), y0);

    // Channel d0+1: taps in s.z/s.w ; weights in wv.z/wv.w ; x in high half of xs
    float y1 = bf16_hi(s.z) * bf16_lo(wv.z);
    y1 = fmaf(bf16_lo(s.w), bf16_hi(wv.z), y1);
    y1 = fmaf(bf16_hi(s.w), bf16_lo(wv.w), y1);
    y1 = fmaf(bf16_hi(xs),  bf16_hi(wv.w), y1);

    y0 = silu(y0);
    y1 = silu(y1);

    // Pack two bf16 results -> one 32-bit store (coalesced).
    const uint32_t packed =
        (uint32_t)f32_to_bf16_bits(y0) | ((uint32_t)f32_to_bf16_bits(y1) << 16);
    *reinterpret_cast<uint32_t*>(out + d0) = packed;
}

extern "C" void kernel_launch(void* const* d_in, const int* in_sizes, int n_in,
                              void* d_out, int out_size, void* d_ws, size_t ws_size,
                              hipStream_t stream) {
    (void)in_sizes; (void)n_in; (void)d_ws; (void)ws_size; (void)out_size;
    const __hip_bfloat16* x  = (const __hip_bfloat16*)d_in[0];  // (1, D)
    const __hip_bfloat16* w  = (const __hip_bfloat16*)d_in[1];  // (D, K)
    const __hip_bfloat16* st = (const __hip_bfloat16*)d_in[2];  // (1, D, K)
    __hip_bfloat16* out = (__hip_bfloat16*)d_out;               // (1, D)

    const int threads = CONV_D / VEC;                 // 5120
    const int grid = (threads + BLOCK - 1) / BLOCK;   // 20 blocks
    conv_step_silu_kernel<<<grid, BLOCK, 0, stream>>>(x, w, st, out);
}